// GraphAttentionEmbedding_11630771438012
// MI455X (gfx1250) — compile-verified
//
#include <hip/hip_runtime.h>
#include <hip/hip_bf16.h>
#include <math.h>

#define N_NODES 50000
#define N_EDGES 500000
#define DIM     128
#define TDIM    100
#define NLAYERS 2

typedef __attribute__((ext_vector_type(2))) float v2f;
typedef __attribute__((ext_vector_type(4))) float v4f;
typedef __attribute__((ext_vector_type(8))) float v8f;

static __device__ __forceinline__ v8f wmma4(v2f a, v2f b, v8f c) {
    // D = A(16x4 f32) * B(4x16 f32) + C(16x16 f32)
    return __builtin_amdgcn_wmma_f32_16x16x4_f32(
        /*neg_a=*/false, a, /*neg_b=*/false, b,
        /*c_mod=*/(short)0, c, /*reuse_a=*/false, /*reuse_b=*/false);
}

// order-preserving float <-> uint for atomic segment-max
static __device__ __forceinline__ unsigned enc_f(float f) {
    unsigned u = __float_as_uint(f);
    return (u & 0x80000000u) ? ~u : (u | 0x80000000u);
}
static __device__ __forceinline__ float dec_f(unsigned u) {
    return __uint_as_float((u & 0x80000000u) ? (u & 0x7FFFFFFFu) : ~u);
}

// ---------------------------------------------------------------------------
// Node GEMMs: Q/K/V/Skip = X @ W + b   (blockIdx.y selects which matrix)
// One wave -> 16 rows x 128 cols, K=128 via 32 x WMMA_F32_16X16X4_F32 per tile
// X and outputs are L2-resident (25.6 MB each): regular (RT) cache policy.
// ---------------------------------------------------------------------------
__global__ void k_node_gemm(const float* __restrict__ X,
                            const float* __restrict__ Wq, const float* __restrict__ Wk,
                            const float* __restrict__ Wv, const float* __restrict__ Ws,
                            const float* __restrict__ bq, const float* __restrict__ bk,
                            const float* __restrict__ bv, const float* __restrict__ bs,
                            float* __restrict__ Q, float* __restrict__ K,
                            float* __restrict__ V, float* __restrict__ S,
                            int ntiles)
{
    const float* W; const float* B; float* O;
    switch (blockIdx.y) {
        case 0:  W = Wq; B = bq; O = Q; break;
        case 1:  W = Wk; B = bk; O = K; break;
        case 2:  W = Wv; B = bv; O = V; break;
        default: W = Ws; B = bs; O = S; break;
    }
    const int wave = threadIdx.x >> 5;
    const int lane = threadIdx.x & 31;
    const int tile = blockIdx.x * (blockDim.x >> 5) + wave;
    if (tile >= ntiles) return;
    const int r  = lane & 15;   // row (A) / col (B,D) within tile
    const int h  = lane >> 4;   // half: selects K pair {0,1} vs {2,3}
    const int m0 = tile * 16;

    v8f acc[8] = {};
    for (int kk = 0; kk < DIM; kk += 4) {
        const int k0 = kk + 2 * h;
        v2f a;
        a.x = X[(size_t)(m0 + r) * DIM + k0];
        a.y = X[(size_t)(m0 + r) * DIM + k0 + 1];
#pragma unroll
        for (int t = 0; t < 8; ++t) {
            const int n = t * 16 + r;
            v2f b;
            b.x = W[(size_t)k0 * DIM + n];
            b.y = W[(size_t)(k0 + 1) * DIM + n];
            acc[t] = wmma4(a, b, acc[t]);
        }
    }
#pragma unroll
    for (int t = 0; t < 8; ++t) {
        const int n = t * 16 + r;
        const float bias = B[n];
#pragma unroll
        for (int vv = 0; vv < 8; ++vv)
            O[(size_t)(m0 + vv + 8 * h) * DIM + n] = acc[t][vv] + bias;
    }
}

// ---------------------------------------------------------------------------
// Edge GEMM with fused TGN time encoding:
//   e_proj = cos(t * time_w + time_b) @ We     [E x 128], K = TDIM = 100
// e_proj is a 256 MB stream: store with non-temporal hint (TH=NT) so it
// doesn't evict the L2-resident node tensors.
// ---------------------------------------------------------------------------
__global__ void k_edge_gemm(const float* __restrict__ et,
                            const float* __restrict__ tw, const float* __restrict__ tb,
                            const float* __restrict__ We,
                            float* __restrict__ EP, int ntiles)
{
    const int wave = threadIdx.x >> 5;
    const int lane = threadIdx.x & 31;
    const int tile = blockIdx.x * (blockDim.x >> 5) + wave;
    if (tile >= ntiles) return;
    const int r  = lane & 15;
    const int h  = lane >> 4;
    const int m0 = tile * 16;

    const float te = et[m0 + r];  // this lane's edge timestamp (A rows = edges)

    v8f acc[8] = {};
    for (int kk = 0; kk < TDIM; kk += 4) {
        const int k0 = kk + 2 * h;
        v2f a;
        a.x = __builtin_cosf(te * tw[k0]     + tb[k0]);
        a.y = __builtin_cosf(te * tw[k0 + 1] + tb[k0 + 1]);
#pragma unroll
        for (int t = 0; t < 8; ++t) {
            const int n = t * 16 + r;
            v2f b;
            b.x = We[(size_t)k0 * DIM + n];
            b.y = We[(size_t)(k0 + 1) * DIM + n];
            acc[t] = wmma4(a, b, acc[t]);
        }
    }
#pragma unroll
    for (int t = 0; t < 8; ++t) {
        const int n = t * 16 + r;
#pragma unroll
        for (int vv = 0; vv < 8; ++vv)
            __builtin_nontemporal_store(acc[t][vv],
                &EP[(size_t)(m0 + vv + 8 * h) * DIM + n]);
    }
}

// ---------------------------------------------------------------------------
// Per-(edge,head) attention logit + atomic segment max over destination nodes.
// 128-bit loads; q/k gathers hit L2, e_proj streamed non-temporally.
// ---------------------------------------------------------------------------
__global__ void k_alpha(const float* __restrict__ Q, const float* __restrict__ K,
                        const float* __restrict__ EP, const int* __restrict__ ei,
                        float* __restrict__ alpha, unsigned* __restrict__ nmax)
{
    const int idx = blockIdx.x * blockDim.x + threadIdx.x;
    if (idx >= 2 * N_EDGES) return;
    const int e  = idx >> 1;
    const int hd = idx & 1;
    const int s  = ei[e];
    const int d  = ei[N_EDGES + e];
    const v4f* qr = (const v4f*)(Q  + (size_t)d * DIM + hd * 64);
    const v4f* kr = (const v4f*)(K  + (size_t)s * DIM + hd * 64);
    const v4f* er = (const v4f*)(EP + (size_t)e * DIM + hd * 64);
    float acc = 0.f;
#pragma unroll
    for (int c = 0; c < 16; ++c) {
        v4f qv = qr[c];
        v4f kv = kr[c];
        v4f ev = __builtin_nontemporal_load(&er[c]);
        acc += qv.x * (kv.x + ev.x) + qv.y * (kv.y + ev.y)
             + qv.z * (kv.z + ev.z) + qv.w * (kv.w + ev.w);
    }
    const float al = acc * 0.125f;  // 1/sqrt(64)
    alpha[idx] = al;
    atomicMax(&nmax[(size_t)d * 2 + hd], enc_f(al));
}

// ---------------------------------------------------------------------------
// exp(alpha - max) and atomic segment sum (softmax denominator)
// ---------------------------------------------------------------------------
__global__ void k_expsum(float* __restrict__ alpha, const unsigned* __restrict__ nmax,
                         float* __restrict__ denom, const int* __restrict__ ei)
{
    const int idx = blockIdx.x * blockDim.x + threadIdx.x;
    if (idx >= 2 * N_EDGES) return;
    const int e  = idx >> 1;
    const int hd = idx & 1;
    const int d  = ei[N_EDGES + e];
    const float m  = dec_f(nmax[(size_t)d * 2 + hd]);
    const float ex = __expf(alpha[idx] - m);
    alpha[idx] = ex;
    atomicAdd(&denom[(size_t)d * 2 + hd], ex);
}

// ---------------------------------------------------------------------------
// Weighted message aggregation: out[dst] += (v[src] + e) * softmax(alpha)
// One thread per (edge, 4 channels): b128 V gather (L2) + b128 NT e load,
// 4 scalar f32 atomic adds into the L2-resident output.
// ---------------------------------------------------------------------------
__global__ void k_message(const float* __restrict__ V, const float* __restrict__ EP,
                          const float* __restrict__ alpha, const float* __restrict__ denom,
                          const int* __restrict__ ei, float* __restrict__ Out)
{
    const int idx = blockIdx.x * blockDim.x + threadIdx.x;
    if (idx >= N_EDGES * (DIM / 4)) return;
    const int e  = idx >> 5;          // 32 v4f chunks per edge
    const int c4 = (idx & 31);        // chunk index, channel = c4*4
    const int hd = c4 >> 4;           // head: channels 0-63 vs 64-127
    const int s  = ei[e];
    const int d  = ei[N_EDGES + e];
    const float a = alpha[(size_t)e * 2 + hd] / denom[(size_t)d * 2 + hd];

    const v4f vv = ((const v4f*)(V + (size_t)s * DIM))[c4];
    const v4f ev = __builtin_nontemporal_load(&((const v4f*)(EP + (size_t)e * DIM))[c4]);

    float* o = Out + (size_t)d * DIM + c4 * 4;
    atomicAdd(&o[0], (vv.x + ev.x) * a);
    atomicAdd(&o[1], (vv.y + ev.y) * a);
    atomicAdd(&o[2], (vv.z + ev.z) * a);
    atomicAdd(&o[3], (vv.w + ev.w) * a);
}

__global__ void k_init_nodes(unsigned* __restrict__ nmax, float* __restrict__ denom)
{
    const int idx = blockIdx.x * blockDim.x + threadIdx.x;
    if (idx >= 2 * N_NODES) return;
    nmax[idx]  = 0x007FFFFFu;  // enc_f(-inf)
    denom[idx] = 0.f;
}

// ---------------------------------------------------------------------------
extern "C" void kernel_launch(void* const* d_in, const int* in_sizes, int n_in,
                              void* d_out, int out_size, void* d_ws, size_t ws_size,
                              hipStream_t stream)
{
    const float* x   = (const float*)d_in[0];
    const int*   ei  = (const int*)  d_in[1];
    const float* et  = (const float*)d_in[2];
    // d_in[3] = msg (unused by reference)
    const float* tw  = (const float*)d_in[4];
    const float* tb  = (const float*)d_in[5];
    const float* Wq  = (const float*)d_in[6];
    const float* bq  = (const float*)d_in[7];
    const float* Wk  = (const float*)d_in[8];
    const float* bk  = (const float*)d_in[9];
    const float* Wv  = (const float*)d_in[10];
    const float* bv  = (const float*)d_in[11];
    const float* We  = (const float*)d_in[12];
    const float* Ws  = (const float*)d_in[13];
    const float* bs  = (const float*)d_in[14];
    float* out = (float*)d_out;

    // workspace carve-up (L2-resident node buffers first, big edge stream last)
    char* p = (char*)d_ws;
    const size_t nodeBytes = (size_t)N_NODES * DIM * sizeof(float);   // 25.6 MB
    float*    q     = (float*)p;                 p += nodeBytes;
    float*    kbuf  = (float*)p;                 p += nodeBytes;
    float*    vbuf  = (float*)p;                 p += nodeBytes;
    float*    xbuf  = (float*)p;                 p += nodeBytes;
    float*    alpha = (float*)p;                 p += (size_t)N_EDGES * 2 * sizeof(float);
    float*    denom = (float*)p;                 p += (size_t)N_NODES * 2 * sizeof(float);
    unsigned* nmax  = (unsigned*)p;              p += (size_t)N_NODES * 2 * sizeof(unsigned);
    float*    eproj = (float*)p;                 // E*128*4 = 256 MB

    const int NTN = N_NODES / 16;   // 3125 node tiles (exact)
    const int NTE = N_EDGES / 16;   // 31250 edge tiles (exact)
    const int WPB = 8;              // waves per 256-thread block

    for (int l = 0; l < NLAYERS; ++l) {
        const float* xin  = (l == 0) ? x    : xbuf;
        float*       xout = (l == 0) ? xbuf : out;   // seeded with skip branch below

        // q/k/v projections + skip (x@Ws+bs) written straight into xout
        k_node_gemm<<<dim3((NTN + WPB - 1) / WPB, 4), 256, 0, stream>>>(
            xin,
            Wq + (size_t)l * DIM * DIM, Wk + (size_t)l * DIM * DIM,
            Wv + (size_t)l * DIM * DIM, Ws + (size_t)l * DIM * DIM,
            bq + (size_t)l * DIM,       bk + (size_t)l * DIM,
            bv + (size_t)l * DIM,       bs + (size_t)l * DIM,
            q, kbuf, vbuf, xout, NTN);

        k_init_nodes<<<(2 * N_NODES + 255) / 256, 256, 0, stream>>>(nmax, denom);

        // edge projection with fused time encoding (NT stores)
        k_edge_gemm<<<(NTE + WPB - 1) / WPB, 256, 0, stream>>>(
            et, tw, tb, We + (size_t)l * TDIM * DIM, eproj, NTE);

        // attention logits + scatter softmax
        k_alpha<<<(2 * N_EDGES + 255) / 256, 256, 0, stream>>>(q, kbuf, eproj, ei, alpha, nmax);
        k_expsum<<<(2 * N_EDGES + 255) / 256, 256, 0, stream>>>(alpha, nmax, denom, ei);

        // weighted aggregation into xout (already holds skip branch)
        k_message<<<(N_EDGES * (DIM / 4) + 255) / 256, 256, 0, stream>>>(
            vbuf, eproj, alpha, denom, ei, xout);
    }
}